// MultiAssetGNN_14980845928619
// MI455X (gfx1250) — compile-verified
//
#include <hip/hip_runtime.h>

// ---------------------------------------------------------------------------
// 2-layer GAT forward (PyG GATConv semantics, add_self_loops=True) + mean pool
// fp32 throughout; GEMMs use V_WMMA_F32_16X16X4_F32 with weights staged into
// LDS via the Tensor Data Mover (tensor_load_to_lds + s_wait_tensorcnt);
// edge softmax/aggregation via L2-resident f32 atomics.
// ---------------------------------------------------------------------------

typedef __attribute__((ext_vector_type(2))) float v2f;
typedef __attribute__((ext_vector_type(8))) float v8f;
typedef __attribute__((ext_vector_type(4))) unsigned int v4u;
typedef __attribute__((ext_vector_type(4))) int v4i;
typedef __attribute__((ext_vector_type(8))) int v8i;

#define NEG_SLOPE 0.2f
#define NGMAX 4096

__device__ __forceinline__ void atomicMaxFloat(float* addr, float v) {
    if (v >= 0.0f) atomicMax((int*)addr, __float_as_int(v));
    else           atomicMin((unsigned int*)addr, __float_as_uint(v));
}

__global__ void fill_f32(float* __restrict__ p, float v, long n) {
    long i = (long)blockIdx.x * blockDim.x + threadIdx.x;
    if (i < n) p[i] = v;
}

// TDM: stage a row-major f32 matrix [K x COLS] from global into LDS.
// D# built per CDNA5 ISA (§8.3 group0, §8.4 group1):
//   group0: [1:0]=count=1, [63:32]=lds_addr, [120:64]=global_addr, [127:126]=type 2
//   group1: data_size=2 (4B), tensor_dim0=COLS, tensor_dim1=K,
//           tile_dim0=COLS, tile_dim1=K, tensor_dim0_stride=COLS,
//           tensor_dim1_stride=K*COLS (unused for 2D tiles)
__device__ __forceinline__ void tdm_stage_weights(const float* W, float* lds,
                                                  unsigned K, unsigned C) {
    const unsigned lds_base = (unsigned)(size_t)(void*)lds;   // low 32b = LDS offset
    const unsigned long long ga = (unsigned long long)(size_t)W;
    v4u g0 = { 1u,                                  // count=1, user mode
               lds_base,
               (unsigned)ga,
               (unsigned)((ga >> 32) & 0x1FFFFFFu) | (2u << 30) };  // type=2
    const unsigned S1 = K * C;
    v8i g1 = { (int)(2u << 16),                                  // data_size=4B
               (int)((C & 0xFFFFu) << 16),                       // tensor_dim0 lo
               (int)((C >> 16) | ((K & 0xFFFFu) << 16)),         // dim0 hi | dim1 lo
               (int)((K >> 16) | ((C & 0xFFFFu) << 16)),         // dim1 hi | tile_dim0
               (int)(K & 0xFFFFu),                               // tile_dim1 (tile_dim2=0)
               (int)C,                                           // dim0_stride lo32
               (int)((S1 & 0xFFFFu) << 16),                      // stride0 hi=0 | stride1 lo
               (int)(S1 >> 16) };                                // stride1 hi
    v4i gz4 = { 0, 0, 0, 0 };
    v8i gz8 = { 0, 0, 0, 0, 0, 0, 0, 0 };
    __builtin_amdgcn_tensor_load_to_lds(g0, g1, gz4, gz4, gz8, 0);
    __builtin_amdgcn_s_wait_tensorcnt(0);
}

// ---------------------------------------------------------------------------
// GEMM1: H[N,128] = X[N,64] @ W[64,128]   (one wave = 16 rows x 128 cols)
// f32 WMMA 16x16x4. A frag: lanes 0-15 row M=l, K=k..k+1; lanes 16-31 K=k+2..3
// B frag mirrors A with columns striped over lanes. C: VGPR i = row i (+8 hi).
// W is staged into LDS once per workgroup via the Tensor Data Mover.
// ---------------------------------------------------------------------------
__global__ void gat_gemm1_wmma(const float* __restrict__ X,
                               const float* __restrict__ W,
                               float* __restrict__ H, int N) {
    __shared__ float lW[64 * 128];                 // 32 KB of the 320 KB WGP LDS
    if ((threadIdx.x >> 5) == 0)                   // one wave issues the DMA
        tdm_stage_weights(W, lW, 64u, 128u);
    __syncthreads();

    const int lane = threadIdx.x & 31;
    const int wave = threadIdx.x >> 5;
    const int tileM = blockIdx.x * 8 + wave;
    const int row0 = tileM << 4;
    if (row0 >= N) return;                         // wave-uniform: EXEC stays all-1
    const int half = lane >> 4;
    const int l15  = lane & 15;
    const int rowA = row0 + l15;
    const bool rok = rowA < N;
    const float* xr = X + (size_t)rowA * 64;

    v8f acc[8] = {};
    for (int k = 0; k < 64; k += 4) {
        const int ka = k + (half << 1);
        v2f a = {};
        if (rok) a = *(const v2f*)(xr + ka);       // global_load_b64
#pragma unroll
        for (int t = 0; t < 8; ++t) {
            const int col = (t << 4) + l15;
            v2f b;
            b.x = lW[ka * 128 + col];
            b.y = lW[(ka + 1) * 128 + col];
            acc[t] = __builtin_amdgcn_wmma_f32_16x16x4_f32(
                false, a, false, b, (short)0, acc[t], false, false);
        }
    }
#pragma unroll
    for (int t = 0; t < 8; ++t) {
        const int col = (t << 4) + l15;
#pragma unroll
        for (int i = 0; i < 8; ++i) {
            const int row = row0 + i + (half << 3);
            if (row < N) H[(size_t)row * 128 + col] = acc[t][i];
        }
    }
}

// GEMM2: H2[N,32] = A[N,128] @ W2[128,32]
__global__ void gat_gemm2_wmma(const float* __restrict__ A,
                               const float* __restrict__ W,
                               float* __restrict__ H, int N) {
    __shared__ float lW[128 * 32];                 // 16 KB
    if ((threadIdx.x >> 5) == 0)
        tdm_stage_weights(W, lW, 128u, 32u);
    __syncthreads();

    const int lane = threadIdx.x & 31;
    const int wave = threadIdx.x >> 5;
    const int tileM = blockIdx.x * 8 + wave;
    const int row0 = tileM << 4;
    if (row0 >= N) return;
    const int half = lane >> 4;
    const int l15  = lane & 15;
    const int rowA = row0 + l15;
    const bool rok = rowA < N;
    const float* ar = A + (size_t)rowA * 128;

    v8f acc[2] = {};
    for (int k = 0; k < 128; k += 4) {
        const int ka = k + (half << 1);
        v2f a = {};
        if (rok) a = *(const v2f*)(ar + ka);
#pragma unroll
        for (int t = 0; t < 2; ++t) {
            const int col = (t << 4) + l15;
            v2f b;
            b.x = lW[ka * 32 + col];
            b.y = lW[(ka + 1) * 32 + col];
            acc[t] = __builtin_amdgcn_wmma_f32_16x16x4_f32(
                false, a, false, b, (short)0, acc[t], false, false);
        }
    }
#pragma unroll
    for (int t = 0; t < 2; ++t) {
        const int col = (t << 4) + l15;
#pragma unroll
        for (int i = 0; i < 8; ++i) {
            const int row = row0 + i + (half << 3);
            if (row < N) H[(size_t)row * 32 + col] = acc[t][i];
        }
    }
}

// alpha_src/alpha_dst per (node, head) for layer 1 (H=4, C=32)
__global__ void alpha1_kernel(const float* __restrict__ H1,
                              const float* __restrict__ a_src,
                              const float* __restrict__ a_dst,
                              float* __restrict__ asrc, float* __restrict__ adst,
                              int N) {
    int tid = blockIdx.x * blockDim.x + threadIdx.x;
    if (tid >= N * 4) return;
    const int n = tid >> 2, h = tid & 3;
    const float* hv = H1 + (size_t)n * 128 + h * 32;
    const float* as = a_src + h * 32;
    const float* ad = a_dst + h * 32;
    float s0 = 0.f, s1 = 0.f;
#pragma unroll 8
    for (int c = 0; c < 32; ++c) { float v = hv[c]; s0 += v * as[c]; s1 += v * ad[c]; }
    asrc[tid] = s0;
    adst[tid] = s1;
}

// alpha per node for layer 2 (H=1, C=32)
__global__ void alpha2_kernel(const float* __restrict__ H2,
                              const float* __restrict__ a_src,
                              const float* __restrict__ a_dst,
                              float* __restrict__ asrc, float* __restrict__ adst,
                              int N) {
    int n = blockIdx.x * blockDim.x + threadIdx.x;
    if (n >= N) return;
    const float* hv = H2 + (size_t)n * 32;
    float s0 = 0.f, s1 = 0.f;
#pragma unroll 8
    for (int c = 0; c < 32; ++c) { float v = hv[c]; s0 += v * a_src[c]; s1 += v * a_dst[c]; }
    asrc[n] = s0;
    adst[n] = s1;
}

// ---------------- layer 1 edge passes (H=4) ----------------
__global__ void edge1_max(const int* __restrict__ ei, int E, int N,
                          const float* __restrict__ asrc, const float* __restrict__ adst,
                          float* __restrict__ m) {
    int tid = blockIdx.x * blockDim.x + threadIdx.x;
    const int ET = E + N;
    if (tid >= ET * 4) return;
    const int e = tid >> 2, h = tid & 3;
    int s, d;
    if (e < E) { s = ei[e]; d = ei[E + e]; } else { s = d = e - E; }
    float l = asrc[s * 4 + h] + adst[d * 4 + h];
    l = l > 0.f ? l : NEG_SLOPE * l;
    atomicMaxFloat(&m[d * 4 + h], l);
}

__global__ void edge1_expsum(const int* __restrict__ ei, int E, int N,
                             const float* __restrict__ asrc, const float* __restrict__ adst,
                             const float* __restrict__ m, float* __restrict__ den,
                             float* __restrict__ alpha) {
    int tid = blockIdx.x * blockDim.x + threadIdx.x;
    const int ET = E + N;
    if (tid >= ET * 4) return;
    const int e = tid >> 2, h = tid & 3;
    int s, d;
    if (e < E) { s = ei[e]; d = ei[E + e]; } else { s = d = e - E; }
    float l = asrc[s * 4 + h] + adst[d * 4 + h];
    l = l > 0.f ? l : NEG_SLOPE * l;
    float ex = __expf(l - m[d * 4 + h]);
    alpha[(size_t)e * 4 + h] = ex;
    atomicAdd(&den[d * 4 + h], ex);
}

__global__ void edge1_aggregate(const int* __restrict__ ei, int E, int N,
                                const float* __restrict__ alpha, const float* __restrict__ den,
                                const float* __restrict__ H1, float* __restrict__ out) {
    long tid = (long)blockIdx.x * blockDim.x + threadIdx.x;
    const int ET = E + N;
    if (tid >= (long)ET * 32) return;
    const int e = (int)(tid >> 5);
    const int r = (int)(tid & 31);
    const int h = r >> 3;
    const int c4 = (r & 7) << 2;
    int s, d;
    if (e < E) { s = ei[e]; d = ei[E + e]; } else { s = d = e - E; }
    const float a = alpha[(size_t)e * 4 + h] / den[d * 4 + h];
    const float4 hv = *(const float4*)(H1 + (size_t)s * 128 + h * 32 + c4);
    float* o = out + (size_t)d * 128 + h * 32 + c4;
    atomicAdd(o + 0, hv.x * a);
    atomicAdd(o + 1, hv.y * a);
    atomicAdd(o + 2, hv.z * a);
    atomicAdd(o + 3, hv.w * a);
}

// out1 = elu(out1 + b1)
__global__ void elu_bias(float* __restrict__ out, const float* __restrict__ b, long n) {
    long i = (long)blockIdx.x * blockDim.x + threadIdx.x;
    if (i >= n) return;
    float v = out[i] + b[i & 127];
    out[i] = v > 0.f ? v : (__expf(v) - 1.0f);
}

// ---------------- layer 2 edge passes (H=1) ----------------
__global__ void edge2_max(const int* __restrict__ ei, int E, int N,
                          const float* __restrict__ asrc, const float* __restrict__ adst,
                          float* __restrict__ m) {
    int tid = blockIdx.x * blockDim.x + threadIdx.x;
    const int ET = E + N;
    if (tid >= ET) return;
    int s, d;
    if (tid < E) { s = ei[tid]; d = ei[E + tid]; } else { s = d = tid - E; }
    float l = asrc[s] + adst[d];
    l = l > 0.f ? l : NEG_SLOPE * l;
    atomicMaxFloat(&m[d], l);
}

__global__ void edge2_expsum(const int* __restrict__ ei, int E, int N,
                             const float* __restrict__ asrc, const float* __restrict__ adst,
                             const float* __restrict__ m, float* __restrict__ den,
                             float* __restrict__ alpha) {
    int tid = blockIdx.x * blockDim.x + threadIdx.x;
    const int ET = E + N;
    if (tid >= ET) return;
    int s, d;
    if (tid < E) { s = ei[tid]; d = ei[E + tid]; } else { s = d = tid - E; }
    float l = asrc[s] + adst[d];
    l = l > 0.f ? l : NEG_SLOPE * l;
    float ex = __expf(l - m[d]);
    alpha[tid] = ex;
    atomicAdd(&den[d], ex);
}

__global__ void edge2_aggregate(const int* __restrict__ ei, int E, int N,
                                const float* __restrict__ alpha, const float* __restrict__ den,
                                const float* __restrict__ H2, float* __restrict__ out) {
    long tid = (long)blockIdx.x * blockDim.x + threadIdx.x;
    const int ET = E + N;
    if (tid >= (long)ET * 8) return;
    const int e = (int)(tid >> 3);
    const int c4 = (int)(tid & 7) << 2;
    int s, d;
    if (e < E) { s = ei[e]; d = ei[E + e]; } else { s = d = e - E; }
    const float a = alpha[e] / den[d];
    const float4 hv = *(const float4*)(H2 + (size_t)s * 32 + c4);
    float* o = out + (size_t)d * 32 + c4;
    atomicAdd(o + 0, hv.x * a);
    atomicAdd(o + 1, hv.y * a);
    atomicAdd(o + 2, hv.z * a);
    atomicAdd(o + 3, hv.w * a);
}

// ---------------- pooling ----------------
__global__ void pool_accum(const float* __restrict__ out2, const float* __restrict__ b2,
                           const int* __restrict__ batch, float* __restrict__ sums,
                           float* __restrict__ counts, int N) {
    long tid = (long)blockIdx.x * blockDim.x + threadIdx.x;
    if (tid >= (long)N * 32) return;
    const int n = (int)(tid >> 5);
    const int c = (int)(tid & 31);
    const int g = batch[n];
    atomicAdd(&sums[(size_t)g * 32 + c], out2[(size_t)n * 32 + c] + b2[c]);
    if (c == 0) atomicAdd(&counts[g], 1.0f);
}

__global__ void pool_div(float* __restrict__ sums, const float* __restrict__ counts,
                         const int* __restrict__ ng_ptr) {
    int tid = blockIdx.x * blockDim.x + threadIdx.x;
    const int ng = *ng_ptr;
    if (tid >= ng * 32) return;
    sums[tid] /= fmaxf(counts[tid >> 5], 1.0f);
}

// ---------------------------------------------------------------------------
extern "C" void kernel_launch(void* const* d_in, const int* in_sizes, int n_in,
                              void* d_out, int out_size, void* d_ws, size_t ws_size,
                              hipStream_t stream) {
    const float* x      = (const float*)d_in[0];
    const int*   ei     = (const int*)d_in[1];
    const int*   batch  = (const int*)d_in[2];
    const int*   ng_ptr = (const int*)d_in[3];
    const float* W1     = (const float*)d_in[4];
    const float* a_src1 = (const float*)d_in[5];
    const float* a_dst1 = (const float*)d_in[6];
    const float* b1     = (const float*)d_in[7];
    const float* W2     = (const float*)d_in[8];
    const float* a_src2 = (const float*)d_in[9];
    const float* a_dst2 = (const float*)d_in[10];
    const float* b2     = (const float*)d_in[11];
    float* out = (float*)d_out;

    const int N = in_sizes[0] / 64;
    const int E = in_sizes[1] / 2;
    const int ET = E + N;

    // workspace carve-up (floats)
    float* ws    = (float*)d_ws;
    float* h1    = ws;                       // N*128
    float* out1  = h1    + (size_t)N * 128;  // N*128
    float* asrc1 = out1  + (size_t)N * 128;  // N*4
    float* adst1 = asrc1 + (size_t)N * 4;
    float* m1    = adst1 + (size_t)N * 4;
    float* den1  = m1    + (size_t)N * 4;
    float* h2    = den1  + (size_t)N * 4;    // N*32
    float* out2  = h2    + (size_t)N * 32;   // N*32
    float* asrc2 = out2  + (size_t)N * 32;   // N
    float* adst2 = asrc2 + (size_t)N;
    float* m2    = adst2 + (size_t)N;
    float* den2  = m2    + (size_t)N;
    float* al1   = den2  + (size_t)N;        // ET*4
    float* al2   = al1   + (size_t)ET * 4;   // ET
    float* cnts  = al2   + (size_t)ET;       // NGMAX

    const int T = 256;
    const float NEGINF = -__builtin_inff();
#define CDIV(a, b) (int)(((long)(a) + (b) - 1) / (b))

    // init accumulators
    fill_f32<<<CDIV((long)N * 128, T), T, 0, stream>>>(out1, 0.f, (long)N * 128);
    fill_f32<<<CDIV(N * 4, T), T, 0, stream>>>(den1, 0.f, (long)N * 4);
    fill_f32<<<CDIV(N * 4, T), T, 0, stream>>>(m1, NEGINF, (long)N * 4);
    fill_f32<<<CDIV((long)N * 32, T), T, 0, stream>>>(out2, 0.f, (long)N * 32);
    fill_f32<<<CDIV(N, T), T, 0, stream>>>(den2, 0.f, (long)N);
    fill_f32<<<CDIV(N, T), T, 0, stream>>>(m2, NEGINF, (long)N);
    fill_f32<<<CDIV(NGMAX, T), T, 0, stream>>>(cnts, 0.f, (long)NGMAX);
    fill_f32<<<CDIV(out_size, T), T, 0, stream>>>(out, 0.f, (long)out_size);

    // layer 1
    gat_gemm1_wmma<<<CDIV(N, 128), 256, 0, stream>>>(x, W1, h1, N);
    alpha1_kernel<<<CDIV(N * 4, T), T, 0, stream>>>(h1, a_src1, a_dst1, asrc1, adst1, N);
    edge1_max<<<CDIV((long)ET * 4, T), T, 0, stream>>>(ei, E, N, asrc1, adst1, m1);
    edge1_expsum<<<CDIV((long)ET * 4, T), T, 0, stream>>>(ei, E, N, asrc1, adst1, m1, den1, al1);
    edge1_aggregate<<<CDIV((long)ET * 32, T), T, 0, stream>>>(ei, E, N, al1, den1, h1, out1);
    elu_bias<<<CDIV((long)N * 128, T), T, 0, stream>>>(out1, b1, (long)N * 128);

    // layer 2
    gat_gemm2_wmma<<<CDIV(N, 128), 256, 0, stream>>>(out1, W2, h2, N);
    alpha2_kernel<<<CDIV(N, T), T, 0, stream>>>(h2, a_src2, a_dst2, asrc2, adst2, N);
    edge2_max<<<CDIV(ET, T), T, 0, stream>>>(ei, E, N, asrc2, adst2, m2);
    edge2_expsum<<<CDIV(ET, T), T, 0, stream>>>(ei, E, N, asrc2, adst2, m2, den2, al2);
    edge2_aggregate<<<CDIV((long)ET * 8, T), T, 0, stream>>>(ei, E, N, al2, den2, h2, out2);

    // mean pool
    pool_accum<<<CDIV((long)N * 32, T), T, 0, stream>>>(out2, b2, batch, out, cnts, N);
    pool_div<<<CDIV(NGMAX * 32, T), T, 0, stream>>>(out, cnts, ng_ptr);
#undef CDIV
}